// TransformerConv_936302871050
// MI455X (gfx1250) — compile-verified
//
#include <hip/hip_runtime.h>

typedef __attribute__((ext_vector_type(2))) float v2f;
typedef __attribute__((ext_vector_type(8))) float v8f;

#define INC 128   // input channels
#define HC  128   // H*C
#define NH  4     // heads

// ---------------------------------------------------------------------------
// Kernel 1: init segment-max (encoded) and denom buffers
// ---------------------------------------------------------------------------
__global__ void init_seg(unsigned int* __restrict__ m_enc,
                         float* __restrict__ denom, int nh) {
    int i = blockIdx.x * blockDim.x + threadIdx.x;
    if (i < nh) { m_enc[i] = 0u; denom[i] = 0.0f; }
}

// ---------------------------------------------------------------------------
// Kernel 2: fused QKV+root GEMM via V_WMMA_F32_16X16X4_F32.
// One wave32 computes one 16x16 output tile; K loop steps by 4.
// blockIdx.y selects which of the 4 weight matrices.
// ---------------------------------------------------------------------------
__global__ __launch_bounds__(256) void gemm_qkvr(
    const float* __restrict__ x,
    const float* __restrict__ Wq, const float* __restrict__ bq, float* __restrict__ oq,
    const float* __restrict__ Wk, const float* __restrict__ bk, float* __restrict__ ok,
    const float* __restrict__ Wv, const float* __restrict__ bv, float* __restrict__ ov,
    const float* __restrict__ Wr, const float* __restrict__ br, float* __restrict__ orr,
    int n_mtiles)
{
    const float* W; const float* bias; float* out;
    switch (blockIdx.y) {
        case 0:  W = Wq; bias = bq; out = oq;  break;
        case 1:  W = Wk; bias = bk; out = ok;  break;
        case 2:  W = Wv; bias = bv; out = ov;  break;
        default: W = Wr; bias = br; out = orr; break;
    }
    const int wave = threadIdx.x >> 5;
    const int lane = threadIdx.x & 31;
    const int tile = blockIdx.x * 8 + wave;
    const int total_tiles = n_mtiles * (HC / 16);
    if (tile >= total_tiles) return;

    const int mtile = tile >> 3;      // HC/16 == 8 N-tiles per M row
    const int ntile = tile & 7;
    const int m0 = mtile * 16;
    const int n0 = ntile * 16;
    const int half = lane >> 4;       // 0: K pair {0,1}; 1: K pair {2,3}
    const int lid  = lane & 15;

    v8f acc = {};
    const float* arow = x + (size_t)(m0 + lid) * INC + 2 * half;
    #pragma unroll 4
    for (int kk = 0; kk < INC; kk += 4) {
        // A fragment: row m0+lid, K = kk + 2*half + {0,1}  (one b64 load)
        v2f a = *(const v2f*)(arow + kk);
        // B fragment: K-rows striped across lanes within each VGPR
        v2f b;
        b.x = W[(size_t)(kk + 2 * half)     * HC + n0 + lid];
        b.y = W[(size_t)(kk + 2 * half + 1) * HC + n0 + lid];
        acc = __builtin_amdgcn_wmma_f32_16x16x4_f32(
            /*neg_a=*/false, a, /*neg_b=*/false, b,
            /*c_mod=*/(short)0, acc, /*reuse_a=*/false, /*reuse_b=*/false);
    }
    const float bb = bias[n0 + lid];
    #pragma unroll
    for (int j = 0; j < 8; ++j) {
        // D layout: VGPR j -> row (m0 + half*8 + j), col n0+lid
        out[(size_t)(m0 + half * 8 + j) * HC + n0 + lid] = acc[j] + bb;
    }
}

// ---------------------------------------------------------------------------
// Kernel 3: per-edge attention logits + segment max (wave32 per edge).
// Lane L loads float4 covering channels [4L,4L+3] => head = L>>3.
// 8-lane xor-shuffle reduction yields per-head dot products.
// ---------------------------------------------------------------------------
__global__ __launch_bounds__(256) void edge_scores(
    const float* __restrict__ q, const float* __restrict__ k,
    const long long* __restrict__ ei, int E,
    float* __restrict__ scores, unsigned int* __restrict__ m_enc)
{
    const int widx = (int)((blockIdx.x * blockDim.x + threadIdx.x) >> 5);
    const int lane = threadIdx.x & 31;
    if (widx >= E) return;
    const long long r = ei[widx];         // source
    const long long c = ei[E + widx];     // destination
    const float4 qa = ((const float4*)(q + (size_t)c * HC))[lane];
    const float4 ka = ((const float4*)(k + (size_t)r * HC))[lane];
    float p = qa.x * ka.x + qa.y * ka.y + qa.z * ka.z + qa.w * ka.w;
    p += __shfl_xor(p, 4, 32);
    p += __shfl_xor(p, 2, 32);
    p += __shfl_xor(p, 1, 32);
    if ((lane & 7) == 0) {
        const int h = lane >> 3;
        const float s = p * 0.17677669529663687f;   // 1/sqrt(32)
        scores[(size_t)widx * NH + h] = s;
        // order-preserving float->uint encoding for integer atomicMax
        unsigned int bits = __float_as_uint(s);
        unsigned int enc = ((int)bits < 0) ? ~bits : (bits | 0x80000000u);
        atomicMax(&m_enc[(size_t)c * NH + h], enc);
    }
}

// ---------------------------------------------------------------------------
// Kernel 4: numerically stable exp + denominator accumulation
// ---------------------------------------------------------------------------
__global__ void edge_exp(const long long* __restrict__ ei, int E,
                         float* __restrict__ scores,
                         const unsigned int* __restrict__ m_enc,
                         float* __restrict__ denom)
{
    const int idx = blockIdx.x * blockDim.x + threadIdx.x;
    if (idx >= E * NH) return;
    const int e = idx >> 2;
    const int h = idx & 3;
    const long long c = ei[E + e];
    const unsigned int u = m_enc[(size_t)c * NH + h];
    const float m = (u & 0x80000000u) ? __uint_as_float(u ^ 0x80000000u)
                                      : __uint_as_float(~u);
    const float ex = __expf(scores[idx] - m);
    scores[idx] = ex;                     // reuse buffer as exp(score - m)
    atomicAdd(&denom[(size_t)c * NH + h], ex);
}

// ---------------------------------------------------------------------------
// Kernel 5: weighted scatter of values (wave32 per edge, float4 per lane)
// ---------------------------------------------------------------------------
__global__ __launch_bounds__(256) void edge_scatter(
    const float* __restrict__ v, const long long* __restrict__ ei, int E,
    const float* __restrict__ ex, const float* __restrict__ denom,
    float* __restrict__ out)
{
    const int widx = (int)((blockIdx.x * blockDim.x + threadIdx.x) >> 5);
    const int lane = threadIdx.x & 31;
    if (widx >= E) return;
    const long long r = ei[widx];
    const long long c = ei[E + widx];
    const int h = lane >> 3;
    const float w = ex[(size_t)widx * NH + h] / denom[(size_t)c * NH + h];
    const float4 va = ((const float4*)(v + (size_t)r * HC))[lane];
    float* o = out + (size_t)c * HC + lane * 4;
    atomicAdd(o + 0, va.x * w);
    atomicAdd(o + 1, va.y * w);
    atomicAdd(o + 2, va.z * w);
    atomicAdd(o + 3, va.w * w);
}

// ---------------------------------------------------------------------------
extern "C" void kernel_launch(void* const* d_in, const int* in_sizes, int n_in,
                              void* d_out, int out_size, void* d_ws, size_t ws_size,
                              hipStream_t stream)
{
    const float*     x  = (const float*)d_in[0];
    const long long* ei = (const long long*)d_in[1];   // int64 in reference
    const float* Wq = (const float*)d_in[2];
    const float* bq = (const float*)d_in[3];
    const float* Wk = (const float*)d_in[4];
    const float* bk = (const float*)d_in[5];
    const float* Wv = (const float*)d_in[6];
    const float* bv = (const float*)d_in[7];
    const float* Wr = (const float*)d_in[8];
    const float* br = (const float*)d_in[9];
    float* out = (float*)d_out;

    const int N = in_sizes[0] / INC;
    const int E = in_sizes[1] / 2;

    // workspace layout
    float* q      = (float*)d_ws;
    float* k      = q + (size_t)N * HC;
    float* v      = k + (size_t)N * HC;
    float* scores = v + (size_t)N * HC;                 // E*NH floats
    unsigned int* m_enc = (unsigned int*)(scores + (size_t)E * NH);
    float* denom  = (float*)(m_enc + (size_t)N * NH);

    // 1) init segment buffers
    {
        int nh = N * NH;
        init_seg<<<(nh + 255) / 256, 256, 0, stream>>>(m_enc, denom, nh);
    }
    // 2) fused WMMA GEMMs: q,k,v -> ws; root -> d_out
    {
        int n_mtiles = N / 16;                 // N = 50000 = 16*3125
        int total_tiles = n_mtiles * (HC / 16);
        dim3 grid((total_tiles + 7) / 8, 4);
        gemm_qkvr<<<grid, 256, 0, stream>>>(x,
                                            Wq, bq, q,
                                            Wk, bk, k,
                                            Wv, bv, v,
                                            Wr, br, out,
                                            n_mtiles);
    }
    // 3) per-edge scores + segment max
    {
        int blocks = (E + 7) / 8;              // 8 wave32 (= 8 edges) per block
        edge_scores<<<blocks, 256, 0, stream>>>(q, k, ei, E, scores, m_enc);
    }
    // 4) exp + denom
    {
        int total = E * NH;
        edge_exp<<<(total + 255) / 256, 256, 0, stream>>>(ei, E, scores, m_enc, denom);
    }
    // 5) weighted value scatter into out (root already there)
    {
        int blocks = (E + 7) / 8;
        edge_scatter<<<blocks, 256, 0, stream>>>(v, ei, E, scores, denom, out);
    }
}